// NPCLoss_56659208569169
// MI455X (gfx1250) — compile-verified
//
#include <hip/hip_runtime.h>
#include <hip/hip_bf16.h>
#include <stdint.h>

#define B_ROWS   8192
#define C_COLS   32000
#define EPS      0.1f
#define TILE     2048
#define NBUF     3
#define THREADS1 256
#define THREADS2 1024

typedef unsigned int v4u __attribute__((ext_vector_type(4)));
typedef int          v8i __attribute__((ext_vector_type(8)));
typedef int          v4i __attribute__((ext_vector_type(4)));

// Build a 1-D TDM descriptor (ISA cdna5 §8) and issue TENSOR_LOAD_TO_LDS.
// D# group0: count=1 | lds_addr | global_addr[56:0] | type=2
// D# group1: data_size=4B, tensor_dim0 = tile_dim0 = nelem, tensor_dim1=1
__device__ __forceinline__ void tdm_load_row_tile(uint32_t lds_byte_addr,
                                                  const float* gsrc,
                                                  uint32_t nelem) {
  uint64_t ga = (uint64_t)(uintptr_t)gsrc;
  v4u g0;
  g0.x = 1u;                                                 // count=1, user mode
  g0.y = lds_byte_addr;                                      // lds_addr (bytes)
  g0.z = (uint32_t)ga;                                       // global_addr[31:0]
  g0.w = (uint32_t)((ga >> 32) & 0x01FFFFFFu) | (2u << 30);  // addr[56:32] | type=2
  v8i g1;
  g1[0] = (int)(2u << 16);                                   // data_size=4B (bits 17:16)
  g1[1] = (int)((nelem & 0xFFFFu) << 16);                    // tensor_dim0[15:0] @79:48
  g1[2] = (int)((nelem >> 16) | (1u << 16));                 // tensor_dim0[31:16], tensor_dim1=1
  g1[3] = (int)((nelem & 0xFFFFu) << 16);                    // tile_dim0 @127:112
  g1[4] = 0;                                                 // tile_dim1/2 = 0 (unused)
  g1[5] = (int)nelem;                                        // tensor_dim0_stride lo
  g1[6] = 0;
  g1[7] = 0;
  v4i gz = {0, 0, 0, 0};
#if __clang_major__ >= 23
  v8i gz8 = {0, 0, 0, 0, 0, 0, 0, 0};
  __builtin_amdgcn_tensor_load_to_lds(g0, g1, gz, gz, gz8, 0);
#else
  __builtin_amdgcn_tensor_load_to_lds(g0, g1, gz, gz, 0);
#endif
}

// Phase 1: one workgroup (8 wave32s) per row; TDM triple-buffered streaming.
__global__ __launch_bounds__(THREADS1)
void npc_row_kernel(const float* __restrict__ out, const int* __restrict__ tgt,
                    float* __restrict__ loss, int* __restrict__ neg_cnt) {
  __shared__ __align__(16) float buf[NBUF][TILE];
  __shared__ float w_m1[8], w_m2[8], w_s[8];

  const int row = blockIdx.x;
  const int tid = threadIdx.x;
  const float* rowp = out + (size_t)row * C_COLS;

  const int T = (C_COLS + TILE - 1) / TILE;   // 16 tiles (15x2048 + 1x1280)
  const bool issuer = (tid < 32);             // wave 0 drives the TDM

  if (issuer) {                               // prologue: 2 tiles in flight
    tdm_load_row_tile((uint32_t)(uintptr_t)&buf[0][0], rowp, TILE);
    tdm_load_row_tile((uint32_t)(uintptr_t)&buf[1][0], rowp + TILE, TILE);
  }

  float m1 = -INFINITY;   // running max (also softmax scale)
  float m2 = -INFINITY;   // running 2nd max (multiset)
  float s  = 0.0f;        // sum exp(x - m1)

  for (int t = 0; t < T; ++t) {
    const int n = (C_COLS - t * TILE < TILE) ? (C_COLS - t * TILE) : TILE;
    if (issuer) {
      if (t + 2 < T) {
        const int rem = C_COLS - (t + 2) * TILE;
        const int nn  = rem < TILE ? rem : TILE;
        tdm_load_row_tile((uint32_t)(uintptr_t)&buf[(t + 2) % NBUF][0],
                          rowp + (size_t)(t + 2) * TILE, (uint32_t)nn);
        __builtin_amdgcn_s_wait_tensorcnt(2);  // tile t done; t+1,t+2 in flight
      } else if (t + 1 < T) {
        __builtin_amdgcn_s_wait_tensorcnt(1);  // tile t done; t+1 in flight
      } else {
        __builtin_amdgcn_s_wait_tensorcnt(0);  // last tile done
      }
    }
    __syncthreads();  // release all waves: tile t is in LDS

    const float4* vb = reinterpret_cast<const float4*>(&buf[t % NBUF][0]);
    const int n4 = n >> 2;   // n is always a multiple of 4 (2048 / 1280)
    for (int i = tid; i < n4; i += THREADS1) {
      float4 v = vb[i];
      #pragma unroll
      for (int c = 0; c < 4; ++c) {
        float x = (c == 0) ? v.x : (c == 1) ? v.y : (c == 2) ? v.z : v.w;
        if (x > m1) {                       // rare once max settles
          s = s * __expf(m1 - x) + 1.0f;
          m2 = m1;
          m1 = x;
        } else {                            // common path: 1 exp
          s += __expf(x - m1);
          if (x > m2) m2 = x;
        }
      }
    }
    __syncthreads();  // everyone done with buf[t%NBUF] before it is re-DMA'd
  }

  // Cross-lane merge of (m1, m2, s) within each wave32.
  for (int off = 16; off > 0; off >>= 1) {
    float om1 = __shfl_xor(m1, off, 32);
    float om2 = __shfl_xor(m2, off, 32);
    float os  = __shfl_xor(s,  off, 32);
    float nm  = fmaxf(m1, om1);
    s  = s * __expf(m1 - nm) + os * __expf(om1 - nm);
    m2 = fmaxf(fminf(m1, om1), fmaxf(m2, om2));
    m1 = nm;
  }
  if ((tid & 31) == 0) {
    const int w = tid >> 5;
    w_m1[w] = m1; w_m2[w] = m2; w_s[w] = s;
  }
  __syncthreads();

  if (tid == 0) {
    float M1 = w_m1[0], M2 = w_m2[0], S = w_s[0];
    for (int w = 1; w < 8; ++w) {
      float am1 = w_m1[w], am2 = w_m2[w], as = w_s[w];
      float nm  = fmaxf(M1, am1);
      S  = S * __expf(M1 - nm) + as * __expf(am1 - nm);
      M2 = fmaxf(fminf(M1, am1), fmaxf(M2, am2));
      M1 = nm;
    }
    const float lse = __logf(S) + M1;
    const float tl  = rowp[tgt[row]];
    const float margin1 = tl - M1;
    const float margin  = (margin1 != 0.0f) ? margin1 : (tl - M2);
    const float lv = (margin >= 0.0f) ? fmaxf(0.0f, 1.0f - margin)
                                      : fmaxf(0.0f, 1.0f - tl + lse);
    loss[row] = lv;
    if (margin < 0.0f) atomicAdd(neg_cnt, 1);
  }
}

// Phase 2: single block. Register-blocked bitonic sort of 8192 floats
// (strides j<8 in VGPRs, j>=8 in LDS), then scan, mask, reduce.
__global__ __launch_bounds__(THREADS2)
void npc_final_kernel(const float* __restrict__ loss, const int* __restrict__ neg_cnt,
                      float* __restrict__ out) {
  __shared__ float sd[B_ROWS];
  __shared__ float aux[THREADS2];
  __shared__ float w_np[32];
  __shared__ int   w_ct[32];
  const int tid  = threadIdx.x;
  const int base = tid * 8;            // each thread owns 8 consecutive elements

  for (int i = tid; i < B_ROWS; i += THREADS2) sd[i] = loss[i];
  __syncthreads();

  float r[8];
  #pragma unroll
  for (int q = 0; q < 8; ++q) r[q] = sd[base + q];

  auto ce = [&](int a, int b, bool up) {   // compare-exchange r[a],r[b]
    const float x = r[a], y = r[b];
    const bool sw = up ? (x > y) : (x < y);
    r[a] = sw ? y : x;
    r[b] = sw ? x : y;
  };

  // Levels k=2,4,8 entirely in registers (direction from global index).
  ce(0,1,true);  ce(2,3,false); ce(4,5,true);  ce(6,7,false);           // k=2 j=1
  ce(0,2,true);  ce(1,3,true);  ce(4,6,false); ce(5,7,false);           // k=4 j=2
  ce(0,1,true);  ce(2,3,true);  ce(4,5,false); ce(6,7,false);           // k=4 j=1
  {
    const bool u = ((base & 8) == 0);                                   // k=8
    ce(0,4,u); ce(1,5,u); ce(2,6,u); ce(3,7,u);                         //   j=4
    ce(0,2,u); ce(1,3,u); ce(4,6,u); ce(5,7,u);                         //   j=2
    ce(0,1,u); ce(2,3,u); ce(4,5,u); ce(6,7,u);                         //   j=1
  }

  // Levels k=16..8192: strides j>=8 via LDS, j=4,2,1 back in registers.
  for (unsigned k = 16; k <= (unsigned)B_ROWS; k <<= 1) {
    #pragma unroll
    for (int q = 0; q < 8; ++q) sd[base + q] = r[q];
    __syncthreads();
    for (unsigned j = k >> 1; j >= 8; j >>= 1) {
      for (unsigned i = tid; i < (unsigned)B_ROWS; i += THREADS2) {
        const unsigned ixj = i ^ j;
        if (ixj > i) {
          const float a = sd[i];
          const float b = sd[ixj];
          const bool up = ((i & k) == 0);
          if (up ? (a > b) : (a < b)) { sd[i] = b; sd[ixj] = a; }
        }
      }
      __syncthreads();
    }
    #pragma unroll
    for (int q = 0; q < 8; ++q) r[q] = sd[base + q];
    const bool u = ((base & k) == 0);    // uniform within the 8-chunk
    ce(0,4,u); ce(1,5,u); ce(2,6,u); ce(3,7,u);
    ce(0,2,u); ce(1,3,u); ce(4,6,u); ce(5,7,u);
    ce(0,1,u); ce(2,3,u); ce(4,5,u); ce(6,7,u);
  }
  // r[] now holds the sorted array segment [base, base+8).

  // Two-level inclusive scan: 8 elements/thread + Hillis-Steele on 1024 partials.
  float cum[8];
  float run = 0.0f;
  #pragma unroll
  for (int q = 0; q < 8; ++q) { run += r[q]; cum[q] = run; }
  aux[tid] = run;
  __syncthreads();
  for (int st = 1; st < THREADS2; st <<= 1) {
    const float a   = aux[tid];
    const float add = (tid >= st) ? aux[tid - st] : 0.0f;
    __syncthreads();
    aux[tid] = a + add;
    __syncthreads();
  }
  const float prefix = (tid > 0) ? aux[tid - 1] : 0.0f;

  const float neg = (float)neg_cnt[0];
  const float thr = (1.0f - EPS) * (1.0f - EPS) * (float)B_ROWS + (1.0f - EPS) * neg;

  float np1 = 0.0f;
  int   ct  = 0;
  #pragma unroll
  for (int q = 0; q < 8; ++q) {
    const float c   = cum[q] + prefix;
    const float idx = (float)(base + q);
    const bool  m   = (c <= thr + 1.0f - idx);
    ct  += m ? 1 : 0;
    np1 += m ? r[q] : 0.0f;
  }
  for (int off = 16; off > 0; off >>= 1) {
    np1 += __shfl_xor(np1, off, 32);
    ct  += __shfl_xor(ct,  off, 32);
  }
  if ((tid & 31) == 0) { w_np[tid >> 5] = np1; w_ct[tid >> 5] = ct; }
  __syncthreads();

  if (tid == 0) {
    float tn = 0.0f; int tc = 0;
    for (int w = 0; w < 32; ++w) { tn += w_np[w]; tc += w_ct[w]; }
    const float count = (float)tc;
    const float np2   = thr - count;
    out[0] = fmaxf(tn, np2) / count;
  }
}

__global__ void npc_init_kernel(int* neg_cnt) {
  if (threadIdx.x == 0) *neg_cnt = 0;
}

extern "C" void kernel_launch(void* const* d_in, const int* in_sizes, int n_in,
                              void* d_out, int out_size, void* d_ws, size_t ws_size,
                              hipStream_t stream) {
  const float* output = (const float*)d_in[0];
  const int*   target = (const int*)d_in[1];
  float* loss = (float*)d_ws;
  int*   neg  = (int*)((char*)d_ws + (size_t)B_ROWS * sizeof(float));
  float* outp = (float*)d_out;

  npc_init_kernel<<<1, 32, 0, stream>>>(neg);
  npc_row_kernel<<<B_ROWS, THREADS1, 0, stream>>>(output, target, loss, neg);
  npc_final_kernel<<<1, THREADS2, 0, stream>>>(loss, neg, outp);
}